// GIKTOld_83279415869897
// MI455X (gfx1250) — compile-verified
//
#include <hip/hip_runtime.h>
#include <math.h>

// Problem constants (from reference)
#define Bq   128
#define Sq   128
#define Dq   100
#define RKq  10
#define KCq  4
#define NEGq (-1000000000.0f)

typedef float v2f __attribute__((ext_vector_type(2)));
typedef float v8f __attribute__((ext_vector_type(8)));

static inline int cdiv(int a, int b) { return (a + b - 1) / b; }

// ---------------------------------------------------------------------------
// Generic fp32 WMMA GEMM:  out[M,N] = act( A[M,K] @ W[N,K]^T + bias[N] )
// A[m,:] = prev[(pidx? pidx[m*pstride] : m), :]
//        + (FAN ? 0.25 * sum_{c<4} nxt[(nidx? nidx[m*4+c] : m*4+c), :] : 0)
// One wave computes one 16x16 tile via V_WMMA_F32_16X16X4_F32.
// K is a template constant so the k-loop fully unrolls into a branch-free
// stream of global_load_b64 + v_wmma. Out-of-range rows/cols are CLAMPED
// (not masked): garbage rows/cols of C are simply never stored.
// ---------------------------------------------------------------------------
template <int K, bool FAN>
__global__ __launch_bounds__(128) void gemm_wmma(
    const float* __restrict__ prev, const int* __restrict__ pidx, int pstride,
    const float* __restrict__ nxt,  const int* __restrict__ nidx,
    const float* __restrict__ W,    const float* __restrict__ bias,
    float* __restrict__ out, int M, int N, int act)
{
  const int lane  = threadIdx.x;                       // 0..31 (wave32)
  const int mt    = blockIdx.y * 4 + threadIdx.y;      // m-tile index
  const int l16   = lane & 15;
  const int khalf = (lane >> 4) * 2;                   // lanes 16..31 hold K+2,K+3
  const int n     = blockIdx.x * 16 + l16;             // B/C column
  const int m     = mt * 16 + l16;                     // A row
  const int mi    = (m < M) ? m : (M - 1);             // clamp (garbage discarded)
  const int ni    = (n < N) ? n : (N - 1);

  // ---- resolve all row base pointers ONCE, outside the k-loop ----
  const int pr = pidx ? pidx[mi * pstride] : mi;
  const float* __restrict__ prow = prev + (size_t)pr * K + khalf;
  const float* __restrict__ wrow = W    + (size_t)ni * K + khalf;
  const float* __restrict__ nr0 = prev;
  const float* __restrict__ nr1 = prev;
  const float* __restrict__ nr2 = prev;
  const float* __restrict__ nr3 = prev;
  if (FAN) {
    const int c0 = nidx ? nidx[mi * 4 + 0] : (mi * 4 + 0);
    const int c1 = nidx ? nidx[mi * 4 + 1] : (mi * 4 + 1);
    const int c2 = nidx ? nidx[mi * 4 + 2] : (mi * 4 + 2);
    const int c3 = nidx ? nidx[mi * 4 + 3] : (mi * 4 + 3);
    nr0 = nxt + (size_t)c0 * K + khalf;
    nr1 = nxt + (size_t)c1 * K + khalf;
    nr2 = nxt + (size_t)c2 * K + khalf;
    nr3 = nxt + (size_t)c3 * K + khalf;
  }

  v8f acc = {0.f, 0.f, 0.f, 0.f, 0.f, 0.f, 0.f, 0.f};

#pragma unroll
  for (int k0 = 0; k0 < K; k0 += 4) {
    v2f a = *(const v2f*)(prow + k0);                  // global_load_b64
    if (FAN) {
      v2f s = *(const v2f*)(nr0 + k0);
      s = s + *(const v2f*)(nr1 + k0);
      s = s + *(const v2f*)(nr2 + k0);
      s = s + *(const v2f*)(nr3 + k0);
      a = a + 0.25f * s;
    }
    v2f bf = *(const v2f*)(wrow + k0);                 // B[k][n] = W[n][k]
    acc = __builtin_amdgcn_wmma_f32_16x16x4_f32(false, a, false, bf,
                                                (short)0, acc, false, false);
  }

  // ---- store C (VGPR i: lanes0-15 -> M=i, lanes16-31 -> M=8+i) ----
  if (n < N) {
    const float bv = bias ? bias[n] : 0.f;
    const int mrow0 = mt * 16 + (lane >> 4) * 8;
#pragma unroll
    for (int i = 0; i < 8; ++i) {
      const int mr = mrow0 + i;
      if (mr < M) {
        float v = acc[i] + bv;
        if (act) v = tanhf(v);
        out[(size_t)mr * N + n] = v;
      }
    }
  }
}

static void launch_gemm(const float* prev, const int* pidx, int pstride,
                        const float* nxt, const int* nidx, int fanout,
                        const float* W, const float* bias, float* out,
                        int M, int N, int K, int act, hipStream_t stream) {
  dim3 grid(cdiv(N, 16), cdiv(M, 64)), block(32, 4);
  if (fanout) {
    gemm_wmma<100, true><<<grid, block, 0, stream>>>(
        prev, pidx, pstride, nxt, nidx, W, bias, out, M, N, act);
  } else if (K == 200) {
    gemm_wmma<200, false><<<grid, block, 0, stream>>>(
        prev, pidx, pstride, nxt, nidx, W, bias, out, M, N, act);
  } else {
    gemm_wmma<100, false><<<grid, block, 0, stream>>>(
        prev, pidx, pstride, nxt, nidx, W, bias, out, M, N, act);
  }
}

// ---------------------------------------------------------------------------
// Utility kernels
// ---------------------------------------------------------------------------
__global__ __launch_bounds__(256) void zero_kernel(float* p, int n) {
  int i = blockIdx.x * blockDim.x + threadIdx.x;
  if (i < n) p[i] = 0.f;
}
__global__ __launch_bounds__(256) void copy_kernel(float* dst, const float* src, int n) {
  int i = blockIdx.x * blockDim.x + threadIdx.x;
  if (i < n) dst[i] = src[i];
}

// Build neighbor index chains for aggregate(): idx1=(B,4) idx2=(B,16) idx3=(B,64)
__global__ __launch_bounds__(128) void build_idx_kernel(
    const int* __restrict__ question, const int* __restrict__ qnb,
    const int* __restrict__ snb, int t,
    int* __restrict__ idx1, int* __restrict__ idx2, int* __restrict__ idx3)
{
  int b = blockIdx.x * blockDim.x + threadIdx.x;
  if (b >= Bq) return;
  int q = question[b * Sq + t];
#pragma unroll
  for (int i = 0; i < 4; ++i) {
    int v1 = qnb[q * 4 + i];
    idx1[b * 4 + i] = v1;
#pragma unroll
    for (int j = 0; j < 4; ++j) {
      int v2 = snb[v1 * 4 + j];
      idx2[b * 16 + i * 4 + j] = v2;
#pragma unroll
      for (int c = 0; c < 4; ++c)
        idx3[b * 64 + i * 16 + j * 4 + c] = qnb[v2 * 4 + c];
    }
  }
}

// X1 = concat( mask? agg : Eq[q_t],  Er[r_t] )  -> (B, 2D)
__global__ __launch_bounds__(256) void build_x1_kernel(
    const float* __restrict__ Eq, const float* __restrict__ Er,
    const float* __restrict__ agg,
    const int* __restrict__ question, const int* __restrict__ response,
    const int* __restrict__ mask, int t, float* __restrict__ X1)
{
  int i = blockIdx.x * blockDim.x + threadIdx.x;
  if (i >= Bq * 2 * Dq) return;
  int b = i / (2 * Dq), d = i % (2 * Dq);
  float v;
  if (d < Dq) {
    if (mask[b * Sq + t] != 0) v = agg[b * Dq + d];
    else { int q = question[b * Sq + t]; v = Eq[(size_t)q * Dq + d]; }
  } else {
    int r = response[b * Sq + t];
    v = Er[r * Dq + (d - Dq)];
  }
  X1[i] = v;
}

__device__ __forceinline__ float sigm(float x) { return 1.f / (1.f + expf(-x)); }

// GRU1 gate: h1 <- gru(gi, gh, h1)   (in place, per-element RMW)
__global__ __launch_bounds__(256) void gru_gate1_kernel(
    const float* __restrict__ gi, const float* __restrict__ gh, float* __restrict__ h)
{
  int i = blockIdx.x * blockDim.x + threadIdx.x;
  if (i >= Bq * Dq) return;
  int b = i / Dq, d = i % Dq;
  const float* gib = gi + b * 3 * Dq;
  const float* ghb = gh + b * 3 * Dq;
  float r = sigm(gib[d] + ghb[d]);
  float z = sigm(gib[Dq + d] + ghb[Dq + d]);
  float nn = tanhf(gib[2 * Dq + d] + r * ghb[2 * Dq + d]);
  h[i] = (1.f - z) * nn + z * h[i];
}

// GRU2 gate: g2 = gru(gi, gh, h2); h2 <- (t==0)? h2 : g2; hist[:,t,:] <- g2 if t>0
__global__ __launch_bounds__(256) void gru_gate2_kernel(
    const float* __restrict__ gi, const float* __restrict__ gh,
    float* __restrict__ h2, float* __restrict__ g2, float* __restrict__ hist, int t)
{
  int i = blockIdx.x * blockDim.x + threadIdx.x;
  if (i >= Bq * Dq) return;
  int b = i / Dq, d = i % Dq;
  const float* gib = gi + b * 3 * Dq;
  const float* ghb = gh + b * 3 * Dq;
  float r = sigm(gib[d] + ghb[d]);
  float z = sigm(gib[Dq + d] + ghb[Dq + d]);
  float nn = tanhf(gib[2 * Dq + d] + r * ghb[2 * Dq + d]);
  float g = (1.f - z) * nn + z * h2[i];
  g2[i] = g;
  if (t != 0) {
    h2[i] = g;
    hist[((size_t)b * Sq + t) * Dq + d] = g;
  }
}

// scores[b,s] = (s < t) ? dot(Eq[question[b,s]], Eq[question[b,t+1]]) : -inf
__global__ __launch_bounds__(128) void scores_kernel(
    const float* __restrict__ Eq, const int* __restrict__ question,
    int t, float* __restrict__ scores)
{
  __shared__ float eqn[Dq];
  int b = blockIdx.x, s = threadIdx.x;
  int qn = question[b * Sq + t + 1];
  if (s < Dq) eqn[s] = Eq[(size_t)qn * Dq + s];
  __syncthreads();
  const float* row = Eq + (size_t)question[b * Sq + s] * Dq;
  float acc = 0.f;
  for (int k = 0; k < Dq; ++k) acc += row[k] * eqn[k];
  scores[b * Sq + s] = (s < t) ? acc : -__builtin_inff();
}

// Serial top-RK per batch row (strict '>' keeps lowest index on ties, like lax.top_k)
__global__ __launch_bounds__(128) void topk_kernel(
    const float* __restrict__ scores, int* __restrict__ sidx)
{
  int b = blockIdx.x * blockDim.x + threadIdx.x;
  if (b >= Bq) return;
  const float* sc = scores + b * Sq;
  unsigned long long u0 = 0ull, u1 = 0ull;
  for (int r = 0; r < RKq; ++r) {
    float best = -__builtin_inff(); int bi = 0;
    for (int s = 0; s < Sq; ++s) {
      bool used = (s < 64) ? ((u0 >> s) & 1ull) : ((u1 >> (s - 64)) & 1ull);
      if (!used && sc[s] > best) { best = sc[s]; bi = s; }
    }
    if (bi < 64) u0 |= (1ull << bi); else u1 |= (1ull << (bi - 64));
    sidx[b * RKq + r] = bi;
  }
}

// states (B,11,D): row0 = g2; rows 1..10 = hist[ (t>RK)? sidx : r-1 ]
__global__ __launch_bounds__(256) void build_states_kernel(
    const float* __restrict__ g2, const float* __restrict__ hist,
    const int* __restrict__ sidx, int t, float* __restrict__ states)
{
  int i = blockIdx.x * blockDim.x + threadIdx.x;
  if (i >= Bq * 11 * Dq) return;
  int d = i % Dq, rb = i / Dq, r = rb % 11, b = rb / 11;
  float v;
  if (r == 0) v = g2[b * Dq + d];
  else {
    int src = (t > RKq) ? sidx[b * RKq + (r - 1)] : (r - 1);
    v = hist[((size_t)b * Sq + src) * Dq + d];
  }
  states[i] = v;
}

// qc (B,5,D): row0 = Eq[q_next]; rows 1..4 = Ec[concept_idx] * concept_mask
__global__ __launch_bounds__(256) void build_qc_kernel(
    const float* __restrict__ Eq, const float* __restrict__ Ec,
    const int* __restrict__ question, const int* __restrict__ cidx,
    const unsigned char* __restrict__ cmask, int t, float* __restrict__ qc)
{
  int i = blockIdx.x * blockDim.x + threadIdx.x;
  if (i >= Bq * 5 * Dq) return;
  int d = i % Dq, rb = i / Dq, j = rb % 5, b = rb / 5;
  int qn = question[b * Sq + t + 1];
  float v;
  if (j == 0) v = Eq[(size_t)qn * Dq + d];
  else {
    int ci = cidx[qn * KCq + (j - 1)];
    v = cmask[qn * KCq + (j - 1)] ? Ec[(size_t)ci * Dq + d] : 0.f;
  }
  qc[i] = v;
}

// Attention predict: per batch b compute p, accumulate yhat[b, col]
__global__ __launch_bounds__(64) void predict_kernel(
    const float* __restrict__ qc, const float* __restrict__ states,
    const float* __restrict__ Qt, const float* __restrict__ Kt,
    const float* __restrict__ W_W, const float* __restrict__ b_W,
    int t, float* __restrict__ yhat)
{
  __shared__ float qv[5], kv[11], og[55], tmp[55], pq[5];
  int b = blockIdx.x, tid = threadIdx.x;
  const float* Wqp = W_W;        // W_W[0, :D]
  const float* Wkp = W_W + Dq;   // W_W[0, D:]
  if (tid < 5) {
    const float* row = Qt + ((size_t)b * 5 + tid) * Dq;
    float a = 0.f;
    for (int k = 0; k < Dq; ++k) a += row[k] * Wqp[k];
    qv[tid] = a;
  } else if (tid < 16) {
    int s = tid - 5;
    const float* row = Kt + ((size_t)b * 11 + s) * Dq;
    float a = 0.f;
    for (int k = 0; k < Dq; ++k) a += row[k] * Wkp[k];
    kv[s] = a;
  }
  __syncthreads();
  if (tid < 55) {
    int q = tid / 11, s = tid % 11;
    const float* qr = qc + ((size_t)b * 5 + q) * Dq;
    const float* sr = states + ((size_t)b * 11 + s) * Dq;
    float a = 0.f;
    for (int k = 0; k < Dq; ++k) a += qr[k] * sr[k];
    og[tid] = a;
    bool valid = (s == 0) || (t > RKq) || ((s - 1) < t);
    tmp[tid] = valid ? (qv[q] + kv[s] + b_W[0]) : NEGq;
  }
  __syncthreads();
  if (tid < 5) {
    float mx = -__builtin_inff();
    for (int s = 0; s < 11; ++s) mx = fmaxf(mx, tmp[tid * 11 + s]);
    float den = 0.f;
    for (int s = 0; s < 11; ++s) den += expf(tmp[tid * 11 + s] - mx);
    float acc = 0.f;
    for (int s = 0; s < 11; ++s) acc += (expf(tmp[tid * 11 + s] - mx) / den) * og[tid * 11 + s];
    pq[tid] = acc;
  }
  __syncthreads();
  if (tid == 0) {
    float p = pq[0] + pq[1] + pq[2] + pq[3] + pq[4];
    int col = (t == 0) ? 0 : (t + 1);
    yhat[b * Sq + col] += p;
  }
}

// ---------------------------------------------------------------------------
extern "C" void kernel_launch(void* const* d_in, const int* in_sizes, int n_in,
                              void* d_out, int out_size, void* d_ws, size_t ws_size,
                              hipStream_t stream) {
  (void)in_sizes; (void)n_in; (void)out_size; (void)ws_size;
  const float* Eq      = (const float*)d_in[0];
  const float* Ec      = (const float*)d_in[1];
  const float* Er      = (const float*)d_in[2];
  const float* W_ih1   = (const float*)d_in[3];
  const float* W_hh1   = (const float*)d_in[4];
  const float* b_ih1   = (const float*)d_in[5];
  const float* b_hh1   = (const float*)d_in[6];
  const float* W_ih2   = (const float*)d_in[7];
  const float* W_hh2   = (const float*)d_in[8];
  const float* b_ih2   = (const float*)d_in[9];
  const float* b_hh2   = (const float*)d_in[10];
  const float* Wagg    = (const float*)d_in[11];   // (3,D,D)
  const float* bagg    = (const float*)d_in[12];   // (3,D)
  const float* W_last  = (const float*)d_in[13];
  const float* b_last  = (const float*)d_in[14];
  const float* W_query = (const float*)d_in[15];
  const float* b_query = (const float*)d_in[16];
  const float* W_key   = (const float*)d_in[17];
  const float* b_key   = (const float*)d_in[18];
  const float* W_W     = (const float*)d_in[19];   // (1,2D)
  const float* b_W     = (const float*)d_in[20];   // (1,)
  const float* h1_init = (const float*)d_in[21];
  const float* h2_init = (const float*)d_in[22];
  const int*   question= (const int*)d_in[23];
  const int*   response= (const int*)d_in[24];
  const int*   maskp   = (const int*)d_in[25];
  const int*   qnb     = (const int*)d_in[26];
  const int*   snb     = (const int*)d_in[27];
  const int*   cidx    = (const int*)d_in[28];
  const unsigned char* cmask = (const unsigned char*)d_in[29];
  float* yhat = (float*)d_out;

  // workspace layout (256B-aligned slabs)
  char* cur = (char*)d_ws;
  auto take = [&](size_t bytes) -> void* {
    void* p = (void*)cur; cur += (bytes + 255) & ~(size_t)255; return p;
  };
  int*   idx1   = (int*)  take(Bq * 4  * sizeof(int));
  int*   idx2   = (int*)  take(Bq * 16 * sizeof(int));
  int*   idx3   = (int*)  take(Bq * 64 * sizeof(int));
  float* E0a    = (float*)take((size_t)Bq * Dq * 4);
  float* E0b    = (float*)take((size_t)Bq * Dq * 4);
  float* E1a    = (float*)take((size_t)Bq * 4 * Dq * 4);
  float* E1b    = (float*)take((size_t)Bq * 4 * Dq * 4);
  float* E2a    = (float*)take((size_t)Bq * 16 * Dq * 4);
  float* X1     = (float*)take((size_t)Bq * 2 * Dq * 4);
  float* gi     = (float*)take((size_t)Bq * 3 * Dq * 4);
  float* gh     = (float*)take((size_t)Bq * 3 * Dq * 4);
  float* h1     = (float*)take((size_t)Bq * Dq * 4);
  float* h2     = (float*)take((size_t)Bq * Dq * 4);
  float* g2     = (float*)take((size_t)Bq * Dq * 4);
  float* hist   = (float*)take((size_t)Bq * Sq * Dq * 4);
  float* scoresb= (float*)take((size_t)Bq * Sq * 4);
  int*   sidx   = (int*)  take((size_t)Bq * RKq * sizeof(int));
  float* states = (float*)take((size_t)Bq * 11 * Dq * 4);
  float* qcb    = (float*)take((size_t)Bq * 5 * Dq * 4);
  float* Ktb    = (float*)take((size_t)Bq * 11 * Dq * 4);
  float* Qtb    = (float*)take((size_t)Bq * 5 * Dq * 4);

  const int D = Dq, D2 = 2 * Dq, D3 = 3 * Dq;
  const int* NOIDX = (const int*)0;
  const float* NOF = (const float*)0;

  // --- init state (every call; harness poisons ws once and never restores) ---
  zero_kernel<<<cdiv(Bq * Sq, 256), 256, 0, stream>>>(yhat, Bq * Sq);
  zero_kernel<<<cdiv(Bq * Sq * Dq, 256), 256, 0, stream>>>(hist, Bq * Sq * Dq);
  copy_kernel<<<cdiv(Bq * Dq, 256), 256, 0, stream>>>(h1, h1_init, Bq * Dq);
  copy_kernel<<<cdiv(Bq * Dq, 256), 256, 0, stream>>>(h2, h2_init, Bq * Dq);

  for (int t = 0; t < Sq - 1; ++t) {
    // ---- aggregate(q_t): 3-hop triangular aggregation, each hop = one WMMA GEMM
    build_idx_kernel<<<1, 128, 0, stream>>>(question, qnb, snb, t, idx1, idx2, idx3);
    // i=0: inputs are raw embedding gathers
    launch_gemm(Eq, question + t, Sq, Ec, idx1, 1, Wagg + 0 * D * D, bagg + 0 * D, E0a, Bq,      D, D, 1, stream);
    launch_gemm(Ec, idx1,         1,  Eq, idx2, 1, Wagg + 1 * D * D, bagg + 1 * D, E1a, Bq * 4,  D, D, 1, stream);
    launch_gemm(Eq, idx2,         1,  Ec, idx3, 1, Wagg + 2 * D * D, bagg + 2 * D, E2a, Bq * 16, D, D, 1, stream);
    // i=1
    launch_gemm(E0a, NOIDX, 1, E1a, NOIDX, 1, Wagg + 0 * D * D, bagg + 0 * D, E0b, Bq,     D, D, 1, stream);
    launch_gemm(E1a, NOIDX, 1, E2a, NOIDX, 1, Wagg + 1 * D * D, bagg + 1 * D, E1b, Bq * 4, D, D, 1, stream);
    // i=2
    launch_gemm(E0b, NOIDX, 1, E1b, NOIDX, 1, Wagg + 0 * D * D, bagg + 0 * D, E0a, Bq, D, D, 1, stream);
    // final: tanh(lin(embs0, W_last, b_last))
    launch_gemm(E0a, NOIDX, 1, NOF, NOIDX, 0, W_last, b_last, E0b, Bq, D, D, 1, stream);

    // ---- GRU1: x = [emb_q, Er[r_t]]
    build_x1_kernel<<<cdiv(Bq * D2, 256), 256, 0, stream>>>(
        Eq, Er, E0b, question, response, maskp, t, X1);
    launch_gemm(X1, NOIDX, 1, NOF, NOIDX, 0, W_ih1, b_ih1, gi, Bq, D3, D2, 0, stream);
    launch_gemm(h1, NOIDX, 1, NOF, NOIDX, 0, W_hh1, b_hh1, gh, Bq, D3, D,  0, stream);
    gru_gate1_kernel<<<cdiv(Bq * D, 256), 256, 0, stream>>>(gi, gh, h1);

    // ---- GRU2
    launch_gemm(h1, NOIDX, 1, NOF, NOIDX, 0, W_ih2, b_ih2, gi, Bq, D3, D, 0, stream);
    launch_gemm(h2, NOIDX, 1, NOF, NOIDX, 0, W_hh2, b_hh2, gh, Bq, D3, D, 0, stream);
    gru_gate2_kernel<<<cdiv(Bq * D, 256), 256, 0, stream>>>(gi, gh, h2, g2, hist, t);

    // ---- history selection + attention inputs
    scores_kernel<<<Bq, 128, 0, stream>>>(Eq, question, t, scoresb);
    topk_kernel<<<1, 128, 0, stream>>>(scoresb, sidx);
    build_states_kernel<<<cdiv(Bq * 11 * D, 256), 256, 0, stream>>>(g2, hist, sidx, t, states);
    build_qc_kernel<<<cdiv(Bq * 5 * D, 256), 256, 0, stream>>>(Eq, Ec, question, cidx, cmask, t, qcb);
    launch_gemm(states, NOIDX, 1, NOF, NOIDX, 0, W_query, b_query, Ktb, Bq * 11, D, D, 1, stream);
    launch_gemm(qcb,    NOIDX, 1, NOF, NOIDX, 0, W_key,   b_key,   Qtb, Bq * 5,  D, D, 1, stream);

    // ---- attention + accumulate yhat
    predict_kernel<<<Bq, 64, 0, stream>>>(qcb, states, Qtb, Ktb, W_W, b_W, t, yhat);
  }
}